// Spline_19602230739910
// MI455X (gfx1250) — compile-verified
//
#include <hip/hip_runtime.h>
#include <math.h>

// SE(3) cubic B-spline pose interpolation — two-phase.
//
// Phase 1 (100k threads): d_i = log(inv(T_i)*T_{i+1}) per knot interval,
// stored as 32-byte rows {rho[3], phi[3], |phi|, pad} in workspace
// (3.2 MB, L2-resident). These deltas depend only on the knot pair, so
// computing them once replaces 6M per-timestamp recomputations.
//
// Phase 2 (2M threads): per timestamp, aligned b128 loads of T0 + 3 delta
// rows, then only the exp/compose chain: 2 hardware trig ops per exp via
// half-angle identities, v_rcp for all divisions, |w*phi| = w*|phi| reuses
// the precomputed angle. Streamed I/O uses NT hints.

namespace {

constexpr float kDT    = 0.1f;
constexpr float kStart = 0.0f;
constexpr float kEps   = 1e-6f;
constexpr float kTiny  = 1e-24f;

struct V3 { float x, y, z; };
struct Q4 { float x, y, z, w; };
struct Pose { V3 t; Q4 q; };

__device__ __forceinline__ float frcp(float x){ return __builtin_amdgcn_rcpf(x); } // v_rcp_f32

__device__ __forceinline__ V3 operator+(V3 a, V3 b){ return {a.x+b.x, a.y+b.y, a.z+b.z}; }
__device__ __forceinline__ V3 operator-(V3 a, V3 b){ return {a.x-b.x, a.y-b.y, a.z-b.z}; }
__device__ __forceinline__ V3 operator*(float s, V3 a){ return {s*a.x, s*a.y, s*a.z}; }
__device__ __forceinline__ float dot3(V3 a, V3 b){ return a.x*b.x + a.y*b.y + a.z*b.z; }
__device__ __forceinline__ V3 cross3(V3 a, V3 b){
  return { a.y*b.z - a.z*b.y,
           a.z*b.x - a.x*b.z,
           a.x*b.y - a.y*b.x };
}

// First-quadrant atan2 (y >= 0, x >= 0): t = min/max, degree-11 odd minimax
// polynomial, reflect about pi/2. Max abs error ~3e-6 rad.
__device__ __forceinline__ float atan2_pos(float y, float x){
  float mn = fminf(y, x);
  float mx = fmaxf(y, x);
  float t  = mn * frcp(mx);
  float t2 = t * t;
  float p  =            -0.01172120f;
  p = fmaf(p, t2,        0.05265332f);
  p = fmaf(p, t2,       -0.11643287f);
  p = fmaf(p, t2,        0.19354346f);
  p = fmaf(p, t2,       -0.33262347f);
  p = fmaf(p, t2,        0.99997726f);
  float r = p * t;
  return (y > x) ? (1.5707963268f - r) : r;
}

__device__ __forceinline__ Q4 quat_conj(Q4 q){ return {-q.x, -q.y, -q.z, q.w}; }

__device__ __forceinline__ Q4 quat_mul(Q4 q, Q4 p){
  Q4 r;
  r.x = q.w*p.x + p.w*q.x + (q.y*p.z - q.z*p.y);
  r.y = q.w*p.y + p.w*q.y + (q.z*p.x - q.x*p.z);
  r.z = q.w*p.z + p.w*q.z + (q.x*p.y - q.y*p.x);
  r.w = q.w*p.w - (q.x*p.x + q.y*p.y + q.z*p.z);
  return r;
}

__device__ __forceinline__ V3 quat_rotate(Q4 q, V3 v){
  V3 qv = {q.x, q.y, q.z};
  V3 t  = 2.0f * cross3(qv, v);
  return v + q.w * t + cross3(qv, t);
}

__device__ __forceinline__ V3 so3_log(Q4 q){
  float s = (q.w < 0.0f) ? -1.0f : 1.0f;   // hemisphere fix: qw >= 0
  V3 qv = {s*q.x, s*q.y, s*q.z};
  float qw = s*q.w;
  float n2 = dot3(qv, qv);
  float n  = __builtin_sqrtf(n2 + kTiny);
  float scale = (n < kEps) ? 2.0f : (2.0f * atan2_pos(n, qw) * frcp(n));
  return scale * qv;
}

// Faithful se3_log of the relative pose inv(A)*B; also returns th = |phi|.
__device__ __forceinline__ void delta_se3_log(Pose A, Pose B,
                                              V3& rho, V3& phi, float& th_out){
  Q4 qc = quat_conj(A.q);
  V3 t  = quat_rotate(qc, B.t - A.t);
  Q4 q  = quat_mul(qc, B.q);

  phi = so3_log(q);
  float th2 = dot3(phi, phi);
  float th  = __builtin_sqrtf(th2 + kTiny);
  float sth = __sinf(th);
  float cth = __cosf(th);
  float denom = 2.0f * (1.0f - cth);
  float c = (th < kEps) ? (1.0f/12.0f)
            : (1.0f - th*sth * frcp(fmaxf(denom, kTiny))) * frcp(fmaxf(th2, kTiny));
  V3 pxt = cross3(phi, t);
  rho = t - 0.5f*pxt + c*cross3(phi, pxt);
  th_out = th;
}

// Fused P * se3_exp([rho, phi]) where th = |phi| is supplied (w-scaled
// precomputed angle). Half-angle identities: sin th = 2 sh ch,
// 1 - cos th = 2 sh^2 -> only 2 hardware trig ops per exp.
__device__ __forceinline__ Pose exp_mul(Pose P, V3 rho, V3 phi, float th){
  float th2 = th * th;
  float sh  = __sinf(0.5f*th);                 // v_sin_f32
  float ch  = __cosf(0.5f*th);                 // v_cos_f32
  float sth = 2.0f * sh * ch;                  // sin(th)
  float omc = 2.0f * sh * sh;                  // 1 - cos(th)

  float a    = (th < kEps) ? 0.5f        : omc        * frcp(fmaxf(th2, kTiny));
  float b    = (th < kEps) ? (1.0f/6.0f) : (th - sth) * frcp(fmaxf(th2*th, kTiny));
  float sfac = (th < kEps) ? (0.5f - th2 * (1.0f/48.0f)) : sh * frcp(th);

  V3 pxr = cross3(phi, rho);
  V3 t   = rho + a*pxr + b*cross3(phi, pxr);
  Q4 q   = { sfac*phi.x, sfac*phi.y, sfac*phi.z, ch };

  return { P.t + quat_rotate(P.q, t), quat_mul(P.q, q) };
}

__device__ __forceinline__ Pose load_pose(const float* __restrict__ p){
  Pose T;
  T.t = { p[0], p[1], p[2] };
  T.q = { p[3], p[4], p[5], p[6] };
  return T;
}

__device__ __forceinline__ V3 load_v3(const float* __restrict__ p){
  return { p[0], p[1], p[2] };
}

} // namespace

// ---------------------------------------------------------------------------
// Phase 1: one thread per knot interval.
// Row layout (32 B, b128-aligned): {rho.x,rho.y,rho.z, phi.x,phi.y,phi.z, th, 0}
// ---------------------------------------------------------------------------
extern "C" __global__ __launch_bounds__(256)
void spline_delta_kernel(const float* __restrict__ knots,
                         float* __restrict__ dws,
                         int num_deltas)
{
  int i = blockIdx.x * blockDim.x + threadIdx.x;
  if (i >= num_deltas) return;

  const float* __restrict__ base = knots + (size_t)i * 7u;
  __builtin_prefetch(base, 0, 3);   // global_prefetch_b8 (amortized, 100k threads)
  Pose A = load_pose(base);
  Pose B = load_pose(base + 7);

  V3 rho, phi; float th;
  delta_se3_log(A, B, rho, phi, th);

  float* d = dws + (size_t)i * 8u;
  d[0] = rho.x; d[1] = rho.y; d[2] = rho.z;
  d[3] = phi.x; d[4] = phi.y; d[5] = phi.z;
  d[6] = th;    d[7] = 0.0f;
}

// ---------------------------------------------------------------------------
// Phase 2: one thread per timestamp; exp/compose chain only.
// ---------------------------------------------------------------------------
extern "C" __global__ __launch_bounds__(256)
void spline_eval_kernel(const float* __restrict__ timestamps,
                        const float* __restrict__ knots,
                        const float* __restrict__ dws,
                        float* __restrict__ out,
                        int n, int K)
{
  int i = blockIdx.x * blockDim.x + threadIdx.x;
  if (i >= n) return;

  float ts = __builtin_nontemporal_load(timestamps + i);

  float t_lo = kStart + kDT;
  float t_hi = kStart + kDT * (float)(K - 1) - kDT;
  ts = fminf(fmaxf(ts, t_lo + 1e-4f), t_hi - 1e-4f);

  // 1.0f/0.1f rounds exactly to 10.0f in fp32 (constant-folded).
  float nt = (ts - kStart) * (1.0f / kDT);
  float fl = floorf(nt);
  float u  = nt - fl;
  int sidx = (int)fl - 1;

  const float* __restrict__ kb = knots + (size_t)sidx * 7u;
  const float* __restrict__ dp = dws   + (size_t)sidx * 8u;

  Pose T0 = load_pose(kb);
  V3 r1 = load_v3(dp +  0), p1 = load_v3(dp +  3); float th1 = dp[ 6];
  V3 r2 = load_v3(dp +  8), p2 = load_v3(dp + 11); float th2 = dp[14];
  V3 r3 = load_v3(dp + 16), p3 = load_v3(dp + 19); float th3 = dp[22];

  float u2 = u*u, u3 = u2*u;
  float w1 = (5.0f + 3.0f*u - 3.0f*u2 +      u3) * (1.0f/6.0f);
  float w2 = (1.0f + 3.0f*u + 3.0f*u2 - 2.0f*u3) * (1.0f/6.0f);
  float w3 = u3 * (1.0f/6.0f);

  Pose pose = exp_mul(T0,   w1*r1, w1*p1, w1*th1);
  pose      = exp_mul(pose, w2*r2, w2*p2, w2*th2);
  pose      = exp_mul(pose, w3*r3, w3*p3, w3*th3);

  float* o = out + (size_t)i * 7u;
  __builtin_nontemporal_store(pose.t.x, o + 0);
  __builtin_nontemporal_store(pose.t.y, o + 1);
  __builtin_nontemporal_store(pose.t.z, o + 2);
  __builtin_nontemporal_store(pose.q.x, o + 3);
  __builtin_nontemporal_store(pose.q.y, o + 4);
  __builtin_nontemporal_store(pose.q.z, o + 5);
  __builtin_nontemporal_store(pose.q.w, o + 6);
}

// ---------------------------------------------------------------------------
// Fallback: monolithic kernel (only if workspace is too small).
// ---------------------------------------------------------------------------
extern "C" __global__ __launch_bounds__(256)
void spline_mono_kernel(const float* __restrict__ timestamps,
                        const float* __restrict__ knots,
                        float* __restrict__ out,
                        int n, int K)
{
  int i = blockIdx.x * blockDim.x + threadIdx.x;
  if (i >= n) return;

  float ts = __builtin_nontemporal_load(timestamps + i);
  float t_lo = kStart + kDT;
  float t_hi = kStart + kDT * (float)(K - 1) - kDT;
  ts = fminf(fmaxf(ts, t_lo + 1e-4f), t_hi - 1e-4f);

  float nt = (ts - kStart) * (1.0f / kDT);
  float fl = floorf(nt);
  float u  = nt - fl;
  int sidx = (int)fl - 1;

  const float* __restrict__ base = knots + (size_t)sidx * 7u;

  Pose T0 = load_pose(base);
  Pose T1 = load_pose(base + 7);
  Pose T2 = load_pose(base + 14);
  Pose T3 = load_pose(base + 21);

  float u2 = u*u, u3 = u2*u;
  float w1 = (5.0f + 3.0f*u - 3.0f*u2 +      u3) * (1.0f/6.0f);
  float w2 = (1.0f + 3.0f*u + 3.0f*u2 - 2.0f*u3) * (1.0f/6.0f);
  float w3 = u3 * (1.0f/6.0f);

  V3 r1, p1, r2, p2, r3, p3;
  float a1, a2, a3;
  delta_se3_log(T0, T1, r1, p1, a1);
  delta_se3_log(T1, T2, r2, p2, a2);
  delta_se3_log(T2, T3, r3, p3, a3);

  Pose pose = exp_mul(T0,   w1*r1, w1*p1, w1*a1);
  pose      = exp_mul(pose, w2*r2, w2*p2, w2*a2);
  pose      = exp_mul(pose, w3*r3, w3*p3, w3*a3);

  float* o = out + (size_t)i * 7u;
  __builtin_nontemporal_store(pose.t.x, o + 0);
  __builtin_nontemporal_store(pose.t.y, o + 1);
  __builtin_nontemporal_store(pose.t.z, o + 2);
  __builtin_nontemporal_store(pose.q.x, o + 3);
  __builtin_nontemporal_store(pose.q.y, o + 4);
  __builtin_nontemporal_store(pose.q.z, o + 5);
  __builtin_nontemporal_store(pose.q.w, o + 6);
}

extern "C" void kernel_launch(void* const* d_in, const int* in_sizes, int n_in,
                              void* d_out, int out_size, void* d_ws, size_t ws_size,
                              hipStream_t stream) {
  const float* timestamps = (const float*)d_in[0];
  const float* knots      = (const float*)d_in[1];
  float* out = (float*)d_out;

  int n = in_sizes[0];          // BATCH timestamps
  int K = in_sizes[1] / 7;      // NUM_KNOTS poses of 7 floats
  int num_deltas = K - 1;

  const int block = 256;        // 8 wave32 waves
  const int grid_n = (n + block - 1) / block;

  size_t ws_needed = (size_t)num_deltas * 8u * sizeof(float);
  if (ws_size >= ws_needed) {
    float* dws = (float*)d_ws;
    const int grid_d = (num_deltas + block - 1) / block;
    spline_delta_kernel<<<grid_d, block, 0, stream>>>(knots, dws, num_deltas);
    spline_eval_kernel<<<grid_n, block, 0, stream>>>(timestamps, knots, dws, out, n, K);
  } else {
    spline_mono_kernel<<<grid_n, block, 0, stream>>>(timestamps, knots, out, n, K);
  }
}